// ResnetBlock_7842610283219
// MI455X (gfx1250) — compile-verified
//
#include <hip/hip_runtime.h>

#define B_ 4
#define N_ 4096
#define C_IN_ 64
#define C_FEAT_ 61
#define FILTERS_ 128
#define K_POINTS_ 15
#define NSAMPLE_ 32
#define C_OUT_ 131
#define ALPHA_ 0.3f
#define EPS_ 1e-3f
#define CPAD_ 64
#define KC_ (K_POINTS_*CPAD_)   /* 960, multiple of 4 */
#define SA_STRIDE_ 964          /* 960+4 dwords: kills LDS bank conflicts on A-frag loads */

typedef __attribute__((ext_vector_type(2))) float v2f;
typedef __attribute__((ext_vector_type(8))) float v8f;

// ---------------------------------------------------------------------------
// Repack weights [15][61][128] -> W3[kgroup=240][f=128][kslot=4]
// where kc = kgroup*4 + kslot, k = kc/64, c = kc%64, zeros for c in 61..63.
// This makes each lane's WMMA B-fragment (two consecutive K values for one
// filter column) an adjacent 8-byte pair -> single global_load_b64.
// ---------------------------------------------------------------------------
__global__ void pad_weights_kernel(const float* __restrict__ w, float* __restrict__ w3) {
    int i = blockIdx.x * blockDim.x + threadIdx.x;
    if (i >= (KC_ / 4) * FILTERS_ * 4) return;
    int kslot  = i & 3;
    int f      = (i >> 2) & (FILTERS_ - 1);
    int kgroup = i >> 9;                 // i / (128*4)
    int kc = kgroup * 4 + kslot;
    int k  = kc / CPAD_;
    int c  = kc % CPAD_;
    w3[i] = (c < C_FEAT_) ? w[(k * C_FEAT_ + c) * FILTERS_ + f] : 0.0f;
}

// ---------------------------------------------------------------------------
// Brute-force KNN: one query point per lane, candidates tiled through LDS.
// Register-resident sorted top-32 (ascending d2, ties by ascending index,
// matching jax.lax.top_k(-d2) semantics).
// ---------------------------------------------------------------------------
__global__ __launch_bounds__(256) void knn_kernel(const float* __restrict__ inp,
                                                  int* __restrict__ idx_out) {
    const int tid = threadIdx.x;
    const int b = blockIdx.x / (N_ / 256);
    const int q = (blockIdx.x % (N_ / 256)) * 256 + tid;
    const float* base = inp + (size_t)b * N_ * C_IN_;

    const float qx = base[q * C_IN_ + 0];
    const float qy = base[q * C_IN_ + 1];
    const float qz = base[q * C_IN_ + 2];
    const float sqn = qx * qx + qy * qy + qz * qz;

    float best_d[NSAMPLE_];
    int   best_i[NSAMPLE_];
#pragma unroll
    for (int t = 0; t < NSAMPLE_; ++t) { best_d[t] = 3.4e38f; best_i[t] = 0; }

    __shared__ float sC[256][4];   // x,y,z,|x|^2 of candidate tile

    for (int tile = 0; tile < N_; tile += 256) {
        __syncthreads();
        {
            int c = tile + tid;
            float x = base[c * C_IN_ + 0];
            float y = base[c * C_IN_ + 1];
            float z = base[c * C_IN_ + 2];
            sC[tid][0] = x; sC[tid][1] = y; sC[tid][2] = z;
            sC[tid][3] = x * x + y * y + z * z;
        }
        __syncthreads();
        for (int j = 0; j < 256; ++j) {
            // same formula as reference: sq_n + sq_m - 2*dot
            float d2 = sqn + sC[j][3]
                     - 2.0f * (qx * sC[j][0] + qy * sC[j][1] + qz * sC[j][2]);
            if (d2 < best_d[NSAMPLE_ - 1]) {
                float d = d2; int id = tile + j;
#pragma unroll
                for (int t = 0; t < NSAMPLE_; ++t) {
                    bool sw = (d < best_d[t]) || (d == best_d[t] && id < best_i[t]);
                    float td = sw ? best_d[t] : d;
                    int   ti = sw ? best_i[t] : id;
                    best_d[t] = sw ? d  : best_d[t];
                    best_i[t] = sw ? id : best_i[t];
                    d = td; id = ti;
                }
            }
        }
    }
    int* o = idx_out + ((size_t)b * N_ + q) * NSAMPLE_;
#pragma unroll
    for (int t = 0; t < NSAMPLE_; ++t) o[t] = best_i[t];
}

// ---------------------------------------------------------------------------
// Fused kernel: gather neighbors, kernel-point weights, weighted = w^T*feat,
// GEMM out = weighted * W via V_WMMA_F32_16X16X4_F32, leaky-ReLU + BN, concat.
// Block = 16 points, 256 threads = 8 waves. Wave w owns filter tile [16w,16w+16).
// ---------------------------------------------------------------------------
__global__ __launch_bounds__(256) void kpconv_kernel(
    const float* __restrict__ inp, const int* __restrict__ nidx,
    const float* __restrict__ w3,  const float* __restrict__ kp,
    const float* __restrict__ gamma, const float* __restrict__ beta,
    const float* __restrict__ mmean, const float* __restrict__ mvar,
    float* __restrict__ out) {

    __shared__ float sA[16][SA_STRIDE_];     // weighted matrix, 16 x 960 (padded)
    __shared__ float sKP[K_POINTS_][3];
    __shared__ float sXYZ[16][3];

    const int tid  = threadIdx.x;
    const int wave = tid >> 5;
    const int lane = tid & 31;
    const int b  = blockIdx.x / (N_ / 16);
    const int n0 = (blockIdx.x % (N_ / 16)) * 16;
    const float* base = inp + (size_t)b * N_ * C_IN_;

    if (tid < K_POINTS_ * 3) sKP[tid / 3][tid % 3] = kp[tid];
    if (tid >= 64 && tid < 64 + 48) {
        int t2 = tid - 64; int p = t2 / 3, d = t2 % 3;
        sXYZ[p][d] = base[(n0 + p) * C_IN_ + d];
    }
    __syncthreads();

    // --- Phase 1: per wave, build A rows for points (wave) and (wave+8) ---
    for (int half = 0; half < 2; ++half) {
        const int p = wave + half * 8;
        const int* ip = nidx + ((size_t)b * N_ + (n0 + p)) * NSAMPLE_;
        int gid = ip[lane];                       // lane <-> neighbor s
        float nx = base[gid * C_IN_ + 0];
        float ny = base[gid * C_IN_ + 1];
        float nz = base[gid * C_IN_ + 2];
        float rx = nx - sXYZ[p][0];
        float ry = ny - sXYZ[p][1];
        float rz = nz - sXYZ[p][2];
        float wk[K_POINTS_];
#pragma unroll
        for (int k = 0; k < K_POINTS_; ++k) {
            float dx = rx - sKP[k][0], dy = ry - sKP[k][1], dz = rz - sKP[k][2];
            float dist = sqrtf(dx * dx + dy * dy + dz * dz + 1e-12f);
            wk[k] = fmaxf(1.0f - dist, 0.0f);     // sigma = KP_EXTEND*RADIUS = 1
        }
        float acc0[K_POINTS_], acc1[K_POINTS_];
#pragma unroll
        for (int k = 0; k < K_POINTS_; ++k) { acc0[k] = 0.f; acc1[k] = 0.f; }
        const int c1 = lane + 32;
        for (int s = 0; s < NSAMPLE_; ++s) {
            int gs = __shfl(gid, s, 32);
            const float* fb = base + gs * C_IN_ + 3;
            float f0 = fb[lane];                               // c = lane (<61)
            float f1 = (c1 < C_FEAT_) ? fb[c1] : 0.0f;         // c = lane+32
#pragma unroll
            for (int k = 0; k < K_POINTS_; ++k) {
                float wv = __shfl(wk[k], s, 32);
                acc0[k] = fmaf(wv, f0, acc0[k]);
                acc1[k] = fmaf(wv, f1, acc1[k]);
            }
        }
#pragma unroll
        for (int k = 0; k < K_POINTS_; ++k) {
            sA[p][k * CPAD_ + lane] = acc0[k];
            sA[p][k * CPAD_ + c1]   = acc1[k];    // zero in the pad region
        }
    }
    __syncthreads();

    // --- Phase 2: GEMM 16x128 = A(16x960) * W(960x128), f32 WMMA k-step 4 ---
    const int  ft    = wave * 16;
    const int  lhalf = lane & 15;
    const bool hi    = lane >= 16;
    // ISA 16x4 f32 A layout: lanes 0-15 hold K0/K1, lanes 16-31 hold K2/K3.
    // Both pairs are consecutive in LDS / in W3 -> single b64 load each.
    const float* ap  = &sA[lhalf][hi ? 2 : 0];
    const float* w3p = w3 + (size_t)(ft + lhalf) * 4 + (hi ? 2 : 0);
    v8f acc = {0.f, 0.f, 0.f, 0.f, 0.f, 0.f, 0.f, 0.f};
#pragma unroll 8
    for (int kb = 0; kb < KC_; kb += 4) {
        v2f a  = *(const v2f*)(ap + kb);                          // ds_load_b64
        v2f bb = *(const v2f*)(w3p + (size_t)(kb >> 2) * (FILTERS_ * 4)); // global_load_b64
        acc = __builtin_amdgcn_wmma_f32_16x16x4_f32(
            /*neg_a=*/false, a, /*neg_b=*/false, bb,
            /*c_mod=*/(short)0, acc, /*reuse_a=*/false, /*reuse_b=*/false);
    }

    // --- Phase 3: epilogue (leaky ReLU + BN), D layout: lane<16 -> M=r, else M=8+r
    const int ch = 3 + ft + lhalf;
    const float g  = gamma[ch], bt = beta[ch], mm = mmean[ch];
    const float inv = rsqrtf(mvar[ch] + EPS_);
#pragma unroll
    for (int r = 0; r < 8; ++r) {
        int m = r + (hi ? 8 : 0);
        float v = acc[r];
        v = (v >= 0.f) ? v : ALPHA_ * v;
        out[((size_t)b * N_ + n0 + m) * C_OUT_ + ch] = (v - mm) * inv * g + bt;
    }
    if (tid < 48) {   // xyz channels 0..2 (BN applied to them too)
        int p = tid / 3, d = tid % 3;
        float v = sXYZ[p][d];
        float y = (v - mmean[d]) * rsqrtf(mvar[d] + EPS_) * gamma[d] + beta[d];
        out[((size_t)b * N_ + n0 + p) * C_OUT_ + d] = y;
    }
}

// ---------------------------------------------------------------------------
extern "C" void kernel_launch(void* const* d_in, const int* in_sizes, int n_in,
                              void* d_out, int out_size, void* d_ws, size_t ws_size,
                              hipStream_t stream) {
    const float* inputs  = (const float*)d_in[0];
    const float* weights = (const float*)d_in[1];
    const float* kpts    = (const float*)d_in[2];
    const float* gamma   = (const float*)d_in[3];
    const float* beta    = (const float*)d_in[4];
    const float* mmean   = (const float*)d_in[5];
    const float* mvar    = (const float*)d_in[6];
    float* out = (float*)d_out;

    int*   idx = (int*)d_ws;                                           // 2 MB
    float* w3  = (float*)((char*)d_ws + (size_t)B_ * N_ * NSAMPLE_ * sizeof(int));

    pad_weights_kernel<<<((KC_ / 4) * FILTERS_ * 4 + 255) / 256, 256, 0, stream>>>(weights, w3);
    knn_kernel<<<B_ * (N_ / 256), 256, 0, stream>>>(inputs, idx);
    kpconv_kernel<<<B_ * (N_ / 16), 256, 0, stream>>>(inputs, idx, w3, kpts,
                                                      gamma, beta, mmean, mvar, out);
}